// Intensive_24137716204126
// MI455X (gfx1250) — compile-verified
//
#include <hip/hip_runtime.h>
#include <math.h>

// Problem constants (from reference): B=32, L=2048, DIM=2048
#define BB   32
#define LL   2048
#define DD   2048
#define WPAD 2064   // padded LDS row stride in floats (2064 % 64 == 16 -> conflict-free)

typedef __attribute__((ext_vector_type(2))) float v2f;
typedef __attribute__((ext_vector_type(8))) float v8f;

// Only check for the builtin in the DEVICE pass: amdgcn builtins are invisible
// to the host pass, which is where the previous guard mis-fired.
#if defined(__HIP_DEVICE_COMPILE__) && !__has_builtin(__builtin_amdgcn_wmma_f32_16x16x4_f32)
#error "device: __builtin_amdgcn_wmma_f32_16x16x4_f32 not available"
#endif

// ---------------------------------------------------------------------------
// Kernel 1: start/end logits via V_WMMA_F32_16X16X4_F32.
//   D = A x B + C, A = qa_w (rows 0..1 real, 2..15 zero), B = 16 data rows.
//   D vgpr0 lanes0-15 = start logits of the 16 rows, vgpr1 = end logits.
// Memory-bound: streams 512 MiB of dropout_states once (~23 us at 23.3 TB/s).
// ---------------------------------------------------------------------------
__global__ __launch_bounds__(256) void qa_logits_wmma(
    const float* __restrict__ X,      // (B*L, DIM) row-major
    const float* __restrict__ qa_w,   // (2, DIM)
    const float* __restrict__ qa_b,   // (2,)
    float* __restrict__ start_out,    // (B*L,)
    float* __restrict__ end_out)      // (B*L,)
{
    __shared__ float w_lds[3 * WPAD]; // rows 0,1 = qa_w; row 2 = zeros
    const int t = threadIdx.x;
    for (int i = t; i < 3 * WPAD; i += 256) {
        int r = i / WPAD;
        int c = i - r * WPAD;
        w_lds[i] = (r < 2 && c < DD) ? qa_w[r * DD + c] : 0.0f;
    }
    __syncthreads();

    const int wave = t >> 5;
    const int lane = t & 31;
    const int n    = lane & 15;          // A: weight row M;  B: data row N
    const int kk   = (lane >> 4) << 1;   // K sub-offset: 0 (lanes 0-15) or 2 (lanes 16-31)
    const int wrow = (n < 2) ? n : 2;    // clamp to the zero row

    const size_t rowBase = (size_t)blockIdx.x * 128 + (size_t)wave * 16;
    const float* __restrict__ bptr = X + (rowBase + (size_t)n) * DD + kk; // data stream
    const float* __restrict__ aptr = &w_lds[wrow * WPAD + kk];            // weights (LDS)

    v8f c = {0.f, 0.f, 0.f, 0.f, 0.f, 0.f, 0.f, 0.f};

    for (int k0 = 0; k0 < DD; k0 += 64) {
        if (k0 + 64 < DD) __builtin_prefetch(bptr + k0 + 64, 0, 1); // global_prefetch_b8
        #pragma unroll
        for (int k = k0; k < k0 + 64; k += 4) {
            v2f a = *(const v2f*)(aptr + k);  // ds_load_b64, 16-bit imm offset
            v2f b = *(const v2f*)(bptr + k);  // global_load_b64, 24-bit imm offset
            c = __builtin_amdgcn_wmma_f32_16x16x4_f32(
                    /*neg_a=*/false, a, /*neg_b=*/false, b,
                    /*c_mod=*/(short)0, c, /*reuse_a=*/false, /*reuse_b=*/false);
        }
    }

    if (lane < 16) {
        size_t r = rowBase + (size_t)lane;
        start_out[r] = c[0] + qa_b[0];   // D[m=0][n=lane]
        end_out[r]   = c[1] + qa_b[1];   // D[m=1][n=lane]
    }
}

// ---------------------------------------------------------------------------
// Block reductions (256 threads)
// ---------------------------------------------------------------------------
__device__ __forceinline__ float block_max(float v, float* red) {
    const int t = threadIdx.x;
    __syncthreads();
    red[t] = v;
    __syncthreads();
    for (int off = 128; off > 0; off >>= 1) {
        if (t < off) red[t] = fmaxf(red[t], red[t + off]);
        __syncthreads();
    }
    float r = red[0];
    __syncthreads();
    return r;
}

__device__ __forceinline__ float block_sum(float v, float* red) {
    const int t = threadIdx.x;
    __syncthreads();
    red[t] = v;
    __syncthreads();
    for (int off = 128; off > 0; off >>= 1) {
        if (t < off) red[t] = red[t] + red[t + off];
        __syncthreads();
    }
    float r = red[0];
    __syncthreads();
    return r;
}

// ---------------------------------------------------------------------------
// Kernel 2: per batch — suffix cummax, score_diff, CE partials, cls dot.
// One block of 256 threads per batch (8 elements per thread).
// ---------------------------------------------------------------------------
__global__ __launch_bounds__(256) void per_batch_kernel(
    const float* __restrict__ start_all, const float* __restrict__ end_all,
    const float* __restrict__ cls, const float* __restrict__ lin_w,
    const float* __restrict__ lin_b,
    const int* __restrict__ ctx_starts, const int* __restrict__ start_pos,
    const int* __restrict__ end_pos,
    float* __restrict__ score_diff, float* __restrict__ ws)
{
    __shared__ float red[256];
    __shared__ float csuf[257];
    const int b = blockIdx.x, t = threadIdx.x;
    const float* sl = start_all + (size_t)b * LL;
    const float* el = end_all   + (size_t)b * LL;

    float s[8], e[8];
    #pragma unroll
    for (int i = 0; i < 8; ++i) { s[i] = sl[t * 8 + i]; e[i] = el[t * 8 + i]; }

    // suffix max of end logits: local scan + chunk-suffix scan
    float sfx[8];
    float m = -INFINITY;
    #pragma unroll
    for (int i = 7; i >= 0; --i) { m = fmaxf(m, e[i]); sfx[i] = m; }
    red[t] = m;
    __syncthreads();
    if (t == 0) {
        float mm = -INFINITY;
        csuf[256] = -INFINITY;
        for (int i = 255; i >= 0; --i) { mm = fmaxf(mm, red[i]); csuf[i] = mm; }
    }
    __syncthreads();
    const float tail = csuf[t + 1];
    const int cs = ctx_starts[b];

    float lh = -INFINITY, lsmax = -INFINITY, lemax = -INFINITY;
    #pragma unroll
    for (int i = 0; i < 8; ++i) {
        float suf  = fmaxf(sfx[i], tail);
        float cand = s[i] + suf;
        int pos = t * 8 + i;
        lh = (pos >= cs) ? fmaxf(lh, cand) : lh;
        lsmax = fmaxf(lsmax, s[i]);
        lemax = fmaxf(lemax, e[i]);
    }
    float score_has = block_max(lh, red);
    float Ms = block_max(lsmax, red);
    float Me = block_max(lemax, red);

    float zs = 0.f, ze = 0.f;
    #pragma unroll
    for (int i = 0; i < 8; ++i) { zs += expf(s[i] - Ms); ze += expf(e[i] - Me); }
    float Zs = block_sum(zs, red);
    float Ze = block_sum(ze, red);

    // intensive head dot: cls[b] . lin_w
    float dp = 0.f;
    #pragma unroll
    for (int i = 0; i < 8; ++i) dp += cls[(size_t)b * DD + t * 8 + i] * lin_w[t * 8 + i];
    float Dx = block_sum(dp, red);

    if (t == 0) {
        float lse_s = Ms + logf(Zs);
        float lse_e = Me + logf(Ze);
        int sp = start_pos[b], ep = end_pos[b];
        int vs = (sp != -100), ve = (ep != -100);
        float nll_s = lse_s - sl[vs ? sp : 0];
        float nll_e = lse_e - el[ve ? ep : 0];
        score_diff[b] = sl[0] + el[0] - score_has;
        ws[b]        = Dx + lin_b[0];   // intensive logit x_b
        ws[32 + b]   = nll_s;
        ws[64 + b]   = nll_e;
        ws[96 + b]   = (float)vs;
        ws[128 + b]  = (float)ve;
    }
}

// ---------------------------------------------------------------------------
// Kernel 3: finalize losses (32 scalars -> 1)
// ---------------------------------------------------------------------------
__global__ void finalize_kernel(const float* __restrict__ ws,
                                const int* __restrict__ is_imp,
                                const float* __restrict__ alpha_w,
                                float* __restrict__ out_total)
{
    if (threadIdx.x == 0 && blockIdx.x == 0) {
        float ns = 0.f, ne = 0.f, cs = 0.f, ce = 0.f, il = 0.f;
        for (int b = 0; b < BB; ++b) {
            ns += ws[32 + b] * ws[96 + b];  cs += ws[96 + b];
            ne += ws[64 + b] * ws[128 + b]; ce += ws[128 + b];
            float x = ws[b];
            float y = (float)is_imp[b];
            float l1pe = log1pf(expf(-fabsf(x)));
            float sp_px = fmaxf(x, 0.f)  + l1pe;  // softplus(x)  = -log_sigmoid(-x)
            float sp_nx = fmaxf(-x, 0.f) + l1pe;  // softplus(-x) = -log_sigmoid(x)
            il += y * sp_nx + (1.f - y) * sp_px;
        }
        float start_loss = ns / fmaxf(cs, 1.f);
        float end_loss   = ne / fmaxf(ce, 1.f);
        float qa_loss = 0.5f * (start_loss + end_loss);
        float intensive_loss = il / (float)BB;
        out_total[0] = alpha_w[0] * qa_loss + alpha_w[1] * intensive_loss;
    }
}

// ---------------------------------------------------------------------------
// Host-side launcher
// ---------------------------------------------------------------------------
extern "C" void kernel_launch(void* const* d_in, const int* in_sizes, int n_in,
                              void* d_out, int out_size, void* d_ws, size_t ws_size,
                              hipStream_t stream) {
    (void)in_sizes; (void)n_in; (void)out_size; (void)ws_size;
    const float* X      = (const float*)d_in[0];  // dropout_states (B,L,DIM)
    const float* cls    = (const float*)d_in[1];  // cls_hidden_state (B,DIM)
    const int*   ctx    = (const int*)  d_in[2];  // context_starts (B,)
    const int*   spos   = (const int*)  d_in[3];  // start_positions (B,)
    const int*   epos   = (const int*)  d_in[4];  // end_positions (B,)
    const int*   isimp  = (const int*)  d_in[5];  // is_impossibles (B,)
    const float* qa_w   = (const float*)d_in[6];  // (2,DIM)
    const float* qa_b   = (const float*)d_in[7];  // (2,)
    const float* lin_w  = (const float*)d_in[8];  // (1,DIM)
    const float* lin_b  = (const float*)d_in[9];  // (1,)
    const float* alpha  = (const float*)d_in[10]; // (1,2)

    float* out        = (float*)d_out;
    float* score_diff = out;                       // 32
    float* start_l    = out + BB;                  // 65536
    float* end_l      = out + BB + BB * LL;        // 65536
    float* total      = out + BB + 2 * BB * LL;    // 1
    float* ws         = (float*)d_ws;              // 160 floats used

    // 65536 rows / 128 rows-per-block = 512 blocks
    qa_logits_wmma<<<512, 256, 0, stream>>>(X, qa_w, qa_b, start_l, end_l);
    per_batch_kernel<<<BB, 256, 0, stream>>>(start_l, end_l, cls, lin_w, lin_b,
                                             ctx, spos, epos, score_diff, ws);
    finalize_kernel<<<1, 64, 0, stream>>>(ws, isimp, alpha, total);
}